// PyramidPooling_16913581212065
// MI455X (gfx1250) — compile-verified
//
#include <hip/hip_runtime.h>

// CDNA5 / gfx1250. wave32. D=A*B+C via V_WMMA_F32_16X16X4_F32 (f32 in/out,
// keeps full f32 precision for means over up to 30000 elements).

typedef __attribute__((ext_vector_type(2))) float v2f;
typedef __attribute__((ext_vector_type(8))) float v8f;

#define NBINS 100
#define SORT_THREADS 128
#define DIM 256
#define OUTC 31          // 1+2+4+8+16 pyramid columns
#define NWAVES 16        // waves per pooling workgroup
#define POOL_THREADS (NWAVES * 32)

// ---------------------------------------------------------------------------
// Kernel 1: deterministic stable counting sort (descending degree) per graph.
// One 128-thread workgroup per graph. Writes order[pos] = global row index.
// ---------------------------------------------------------------------------
__global__ __launch_bounds__(SORT_THREADS)
void pp_sort_kernel(const int* __restrict__ counts,
                    const int* __restrict__ degrees,
                    int* __restrict__ order) {
  __shared__ int lhist[NBINS][SORT_THREADS]; // [bin][thread] local hist / cursor
  __shared__ int binBase[NBINS];
  __shared__ int binTotal[NBINS];

  const int g = blockIdx.x;
  int s = 0;
  for (int i = 0; i < g; ++i) s += counts[i];
  const int n = counts[g];
  const int t = threadIdx.x;

  for (int b = 0; b < NBINS; ++b) lhist[b][t] = 0;
  __syncthreads();

  const int chunk = (n + SORT_THREADS - 1) / SORT_THREADS;
  const int i0 = t * chunk;
  const int i1 = (i0 + chunk < n) ? (i0 + chunk) : n;

  for (int i = i0; i < i1; ++i) {
    int d = degrees[s + i];
    d = d < 0 ? 0 : (d >= NBINS ? NBINS - 1 : d);
    lhist[d][t] += 1; // column t owned exclusively by thread t: no atomics
  }
  __syncthreads();

  // exclusive prefix over threads within each bin (stable within bin)
  if (t < NBINS) {
    int running = 0;
    for (int tt = 0; tt < SORT_THREADS; ++tt) {
      int tmp = lhist[t][tt];
      lhist[t][tt] = running;
      running += tmp;
    }
    binTotal[t] = running;
  }
  __syncthreads();

  // descending-degree bin bases (degree 99 first) -> argsort(-deg, stable)
  if (t == 0) {
    int running = 0;
    for (int b = NBINS - 1; b >= 0; --b) {
      binBase[b] = running;
      running += binTotal[b];
    }
  }
  __syncthreads();

  for (int b = 0; b < NBINS; ++b) lhist[b][t] += binBase[b];
  __syncthreads();

  for (int i = i0; i < i1; ++i) {
    int d = degrees[s + i];
    d = d < 0 ? 0 : (d >= NBINS ? NBINS - 1 : d);
    int p = lhist[d][t]++;       // exclusive per (bin, thread): deterministic
    order[s + p] = s + i;        // global row index at sorted position p
  }
}

// map pyramid output column c (0..30) -> [lo,hi) position range + 1/kernel
__device__ __forceinline__ void pp_seg_params(int c, int n,
                                              int& lo, int& hi, float& inv) {
  int p, off;
  if (c == 0)       { p = 1;  off = 0;  }
  else if (c < 3)   { p = 2;  off = 1;  }
  else if (c < 7)   { p = 4;  off = 3;  }
  else if (c < 15)  { p = 8;  off = 7;  }
  else if (c < OUTC){ p = 16; off = 15; }
  else { lo = 0; hi = 0; inv = 0.0f; return; }   // pad column 31
  const int k = (n + p - 1) / p;
  const int j = c - off;
  lo = j * k;
  hi = (lo + k < n) ? (lo + k) : n;  // mean still divides by full k (zero pad)
  inv = 1.0f / (float)k;
}

__device__ __forceinline__ int imin(int a, int b) { return a < b ? a : b; }

// ---------------------------------------------------------------------------
// Kernel 2: pyramid pooling as segmented matmul on the WMMA pipe.
// grid.x = ngraphs*16 : one workgroup per (graph, 16-feature tile).
// 16 waves split the position axis (sub-ranges rounded to 8 positions);
// fixed-order LDS reduction => deterministic float sums.
// ---------------------------------------------------------------------------
__global__ __launch_bounds__(POOL_THREADS)
void pp_pool_kernel(const float* __restrict__ x,
                    const int* __restrict__ counts,
                    const int* __restrict__ order,
                    float* __restrict__ out) {
  __shared__ float red[NWAVES][32][16]; // 32 KB partials

  const int g    = blockIdx.x >> 4;
  const int tile = blockIdx.x & 15;
  const int dd_base = tile * 16;
  const int wave = threadIdx.x >> 5;
  const int lane = threadIdx.x & 31;
  const int col   = lane & 15;       // A: M row; B/C: N col
  const int khalf = (lane >> 4) << 1;// 0 => K{0,1}, 2 => K{2,3}
  const int dcol  = dd_base + col;   // this lane's feature column

  int s = 0;
  for (int i = 0; i < g; ++i) s += counts[i];
  const int n = counts[g];

  // per-lane segment windows for the two 16-column B sets
  int lo0, hi0, lo1, hi1;
  float inv0, inv1;
  pp_seg_params(col,      n, lo0, hi0, inv0);
  pp_seg_params(col + 16, n, lo1, hi1, inv1);

  // position sub-range for this wave, rounded to a multiple of 8 so the
  // unrolled body never crosses into another wave's range; positions >= n
  // carry zero weight, so over-run past n is harmless.
  const int chunks = (n + 3) >> 2;
  int span = ((chunks + NWAVES - 1) / NWAVES) << 2;
  span = (span + 7) & ~7;
  const int wbeg = wave * span;
  const int wend = imin(wbeg + span, n);

  v8f acc0 = {};
  v8f acc1 = {};

  for (int pos = wbeg; pos < wend; pos += 8) {
    // ---- chunk A: positions pos..pos+3, chunk B: pos+4..pos+7 ----
    const int pA0 = pos + khalf,     pA1 = pA0 + 1;
    const int pB0 = pos + 4 + khalf, pB1 = pB0 + 1;

    // clamped gathers (weights are 0 for positions >= n)
    const int rA0 = order[s + imin(pA0, n - 1)];
    const int rA1 = order[s + imin(pA1, n - 1)];
    const int rB0 = order[s + imin(pB0, n - 1)];
    const int rB1 = order[s + imin(pB1, n - 1)];

    v2f aA, aB;                       // A[col][k] = x_sorted[pos+k][dcol]
    aA.x = x[rA0 * DIM + dcol];
    aA.y = x[rA1 * DIM + dcol];
    aB.x = x[rB0 * DIM + dcol];
    aB.y = x[rB1 * DIM + dcol];

    // unconditional clamped prefetch ~96 positions ahead (global_prefetch_b8)
    {
      const int pf  = imin(pos + 96 + khalf, n - 1);
      const int rpf = order[s + pf];
      __builtin_prefetch(x + rpf * DIM + dcol, 0, 1);
    }

    v2f bA0, bA1, bB0, bB1;           // B[k][col] = weight(pos+k, c)
    bA0.x = (pA0 >= lo0 && pA0 < hi0) ? inv0 : 0.0f;
    bA0.y = (pA1 >= lo0 && pA1 < hi0) ? inv0 : 0.0f;
    bA1.x = (pA0 >= lo1 && pA0 < hi1) ? inv1 : 0.0f;
    bA1.y = (pA1 >= lo1 && pA1 < hi1) ? inv1 : 0.0f;
    bB0.x = (pB0 >= lo0 && pB0 < hi0) ? inv0 : 0.0f;
    bB0.y = (pB1 >= lo0 && pB1 < hi0) ? inv0 : 0.0f;
    bB1.x = (pB0 >= lo1 && pB0 < hi1) ? inv1 : 0.0f;
    bB1.y = (pB1 >= lo1 && pB1 < hi1) ? inv1 : 0.0f;

    acc0 = __builtin_amdgcn_wmma_f32_16x16x4_f32(
        false, aA, false, bA0, (short)0, acc0, false, false);
    acc1 = __builtin_amdgcn_wmma_f32_16x16x4_f32(
        false, aA, false, bA1, (short)0, acc1, false, false);
    acc0 = __builtin_amdgcn_wmma_f32_16x16x4_f32(
        false, aB, false, bB0, (short)0, acc0, false, false);
    acc1 = __builtin_amdgcn_wmma_f32_16x16x4_f32(
        false, aB, false, bB1, (short)0, acc1, false, false);
  }

  // stash partials: slot r 0..7 -> acc0[r], 8..15 -> acc1[r-8]
  for (int r = 0; r < 8; ++r) {
    red[wave][lane][r]     = acc0[r];
    red[wave][lane][r + 8] = acc1[r];
  }
  __syncthreads();

  // fixed-order cross-wave reduction: 512 slots over 512 threads
  {
    const int q  = threadIdx.x;      // 0..511
    const int lq = q >> 4;           // lane that produced this slot
    const int r  = q & 15;
    float v = 0.0f;
    for (int w = 0; w < NWAVES; ++w) v += red[w][lq][r]; // deterministic order
    const int colq = lq & 15;
    const int mofs = (lq >= 16) ? 8 : 0;
    int dd, c;
    if (r < 8) { dd = dd_base + r + mofs;        c = colq;      }
    else       { dd = dd_base + (r - 8) + mofs;  c = colq + 16; }
    if (c < OUTC)
      out[g * (DIM * OUTC) + dd * OUTC + c] = v;
  }
}

// ---------------------------------------------------------------------------
extern "C" void kernel_launch(void* const* d_in, const int* in_sizes, int n_in,
                              void* d_out, int out_size, void* d_ws, size_t ws_size,
                              hipStream_t stream) {
  const float* x       = (const float*)d_in[0];
  const int*   counts  = (const int*)d_in[1];
  const int*   degrees = (const int*)d_in[2];
  float*       out     = (float*)d_out;
  int*         order   = (int*)d_ws;          // in_sizes[2] ints of scratch

  const int ngraphs = in_sizes[1];

  pp_sort_kernel<<<ngraphs, SORT_THREADS, 0, stream>>>(counts, degrees, order);
  pp_pool_kernel<<<ngraphs * 16, POOL_THREADS, 0, stream>>>(x, counts, order, out);
}